// PreprocessLayer_47270410060324
// MI455X (gfx1250) — compile-verified
//
#include <hip/hip_runtime.h>
#include <stdint.h>
#include <stddef.h>

// ---------------------------------------------------------------------------
// PreprocessLayer masking for MI455X (gfx1250, wave32).
// data[B=32,N=20000,D=64] f32 -> ((1-mask)*data, mask), both [B,N,D].
// Memory-bound: ~492MB total traffic => ~21us floor @ 23.3 TB/s. No matmul,
// so no WMMA; instead exploit 192MB L2 (warm input, NT output stores),
// wave32 ballot, LDS radix-select, and the gfx1250 async-to-LDS path.
// ---------------------------------------------------------------------------

typedef float v4f __attribute__((ext_vector_type(4)));

#define BATCH 32
#define NVEC  20000
#define V4_PER_VEC 16                      // D=64 floats = 16 float4
#define V4_PER_ROW (NVEC * V4_PER_VEC)     // 320000
#define WORDS_PER_ROW 625                  // 20000 / 32

// workspace layout (bytes)
#define WS_COUNTS 0                        // u32[32]
#define WS_BITS   128                      // u32[32*625] non-empty bitmask
#define WS_THRESH 81920                    // u64[64]: T1[32] then T2[32]

// ---------------- counter-based PRNG (stands in for jax threefry) ----------
__device__ __forceinline__ unsigned pcg(unsigned x) {
  x = x * 747796405u + 2891336453u;
  unsigned w = ((x >> ((x >> 28) + 4u)) ^ x) * 277803737u;
  return (w >> 22) ^ w;
}
__device__ __forceinline__ unsigned rng32(unsigned seed, unsigned strm, unsigned idx) {
  return pcg(idx ^ pcg(seed ^ (strm * 0x9E3779B9u)));
}
// 64-bit sort key = (score << 32) | index : unique per row, ordering identical
// to a stable argsort of the scores (ties broken by index).
__device__ __forceinline__ unsigned long long mkkey1(unsigned seed, int b, int n, bool ne) {
  unsigned h;
  if (ne) {
    h = rng32(seed, 1u, (unsigned)(b * NVEC + n));
    if (h > 0xFFFFFFFEu) h = 0xFFFFFFFEu;       // keep below the "empty" score
  } else {
    h = 0xFFFFFFFFu;                            // empties sort last (r = 2.0)
  }
  return ((unsigned long long)h << 32) | (unsigned)n;
}
__device__ __forceinline__ unsigned long long mkkey2(unsigned seed, int b, int n) {
  unsigned h = rng32(seed, 2u, (unsigned)(b * NVEC + n));
  return ((unsigned long long)h << 32) | (unsigned)n;
}

// ---------------- kernel 0: zero the per-batch counters --------------------
__global__ void k_init(unsigned* __restrict__ counts) {
  if (threadIdx.x < BATCH) counts[threadIdx.x] = 0u;
}

// ---------------- kernel 1: non-empty bitmask + per-row counts -------------
// grid(20, 32) x 1024 threads: one pass over data (164MB, RT hint -> warms L2).
__global__ __launch_bounds__(1024)
void k_count(const v4f* __restrict__ data,
             unsigned* __restrict__ counts,
             unsigned* __restrict__ bits) {
  __shared__ unsigned bsum;
  if (threadIdx.x == 0) bsum = 0u;
  __syncthreads();

  const int b = blockIdx.y;
  const int n = blockIdx.x * 1024 + threadIdx.x;
  const unsigned lane = threadIdx.x & 31u;

  if (n < NVEC) {
    const v4f* p = data + ((size_t)b * NVEC + n) * V4_PER_VEC;
    bool nz = false;
#pragma unroll
    for (int j = 0; j < V4_PER_VEC; ++j) {
      v4f v = p[j];
      nz = nz | (v.x != 0.0f) | (v.y != 0.0f) | (v.z != 0.0f) | (v.w != 0.0f);
    }
    // wave32 ballot == exactly one bitmask word (N is a multiple of 32, so
    // every in-range wave is fully populated).
    unsigned bal = __builtin_amdgcn_ballot_w32(nz);
    if (lane == 0u) {
      bits[b * WORDS_PER_ROW + (n >> 5)] = bal;
      atomicAdd(&bsum, (unsigned)__popc((int)bal));
    }
  }
  __syncthreads();
  if (threadIdx.x == 0) atomicAdd(&counts[b], bsum);
}

// ---------------- kernel 2: MSD radix select of the rank-k key -------------
// grid(64): blockIdx = b*2 + which. One 256-thread workgroup per (row, array).
// Finds T = key with exactly k keys strictly below it, so sel <=> key < T.
__global__ __launch_bounds__(256)
void k_select(const unsigned* __restrict__ counts,
              const unsigned* __restrict__ bits,
              unsigned long long* __restrict__ thresh,
              const int* __restrict__ seedp) {
  const unsigned seed = (unsigned)seedp[0];
  const int b = blockIdx.x >> 1;
  const int which = blockIdx.x & 1;

  const unsigned cnt = counts[b];
  const int k = (which == 0)
                    ? (int)((float)cnt * 0.15f)                 // .astype(int32)
                    : (int)((float)(NVEC - (int)cnt) * 0.1f);
  if (k <= 0) {
    if (threadIdx.x == 0) thresh[which * BATCH + b] = 0ull;     // select none
    return;
  }

  __shared__ unsigned hist[256];
  __shared__ unsigned long long s_prefix;
  __shared__ int s_want;
  if (threadIdx.x == 0) { s_prefix = 0ull; s_want = k; }

  for (int shift = 56; shift >= 0; shift -= 8) {
    hist[threadIdx.x] = 0u;
    __syncthreads();
    const unsigned long long prefix = s_prefix;
    const unsigned long long fixedmask =
        (shift == 56) ? 0ull : (~0ull << (shift + 8));
    for (int n = threadIdx.x; n < NVEC; n += 256) {
      const bool ne = (bits[b * WORDS_PER_ROW + (n >> 5)] >> (n & 31)) & 1u;
      const unsigned long long key =
          (which == 0) ? mkkey1(seed, b, n, ne) : mkkey2(seed, b, n);
      if (((key ^ prefix) & fixedmask) == 0ull)
        atomicAdd(&hist[(unsigned)(key >> shift) & 255u], 1u);
    }
    __syncthreads();
    if (threadIdx.x == 0) {
      int want = s_want;
      unsigned cum = 0u;
      int d = 0;
      for (; d < 256; ++d) {
        if (cum + hist[d] > (unsigned)want) break;
        cum += hist[d];
      }
      if (d > 255) d = 255;  // paranoia guard
      s_prefix = prefix | ((unsigned long long)(unsigned)d << shift);
      s_want = want - (int)cum;
    }
    __syncthreads();
  }
  if (threadIdx.x == 0) thresh[which * BATCH + b] = s_prefix;
}

// ---------------- kernel 3: fused mask/apply streaming kernel --------------
// grid(1250, 32) x 256. Each block streams one contiguous 4KB tile of data
// through LDS via the gfx1250 async-to-LDS path (ASYNCcnt), then emits both
// outputs with non-temporal B128 stores so the 328MB output stream does not
// evict the (L2-resident) input.
__global__ __launch_bounds__(256)
void k_apply(const v4f* __restrict__ data,
             v4f* __restrict__ out,
             const unsigned* __restrict__ bits,
             const unsigned long long* __restrict__ thresh,
             const int* __restrict__ seedp) {
  const unsigned seed = (unsigned)seedp[0];
  const int b = blockIdx.y;
  const int g = blockIdx.x * 256 + threadIdx.x;   // float4 index within row
  const int n = g >> 4;                           // vector index (0..19999)

  const bool ne = (bits[b * WORDS_PER_ROW + (n >> 5)] >> (n & 31)) & 1u;
  const unsigned long long T1 = thresh[b];
  const unsigned long long T2 = thresh[BATCH + b];
  const bool sel = (mkkey1(seed, b, n, ne) < T1) | (mkkey2(seed, b, n) < T2);

  // --- async global -> LDS staging (per-lane B128, wave-wide 512B) ---------
  __shared__ v4f tile[256];
  const size_t off = (size_t)b * V4_PER_ROW + g;
  const v4f* gp = data + off;
  v4f* tp = &tile[threadIdx.x];
  unsigned lds_addr = (unsigned)(size_t)tp;       // addrspace(3) byte offset
  asm volatile("global_load_async_to_lds_b128 %0, %1, off"
               :
               : "v"(lds_addr), "v"(gp)
               : "memory");
  asm volatile("s_wait_asynccnt 0" ::: "memory");
  v4f d = *tp;                                    // ds_load_b128 of own slot

  const float m = sel ? 1.0f : 0.0f;
  v4f zero = {0.0f, 0.0f, 0.0f, 0.0f};
  v4f o1 = sel ? zero : d;                        // (1-mask)*data
  v4f o2 = {m, m, m, m};                          // mask broadcast over D

  __builtin_nontemporal_store(o1, out + off);
  __builtin_nontemporal_store(o2, out + (size_t)BATCH * V4_PER_ROW + off);
}

// ---------------------------------------------------------------------------
extern "C" void kernel_launch(void* const* d_in, const int* in_sizes, int n_in,
                              void* d_out, int out_size, void* d_ws, size_t ws_size,
                              hipStream_t stream) {
  (void)in_sizes; (void)n_in; (void)out_size; (void)ws_size;

  const v4f* data = (const v4f*)d_in[0];
  const int* seedp = (const int*)d_in[1];

  unsigned* counts = (unsigned*)((char*)d_ws + WS_COUNTS);
  unsigned* bits   = (unsigned*)((char*)d_ws + WS_BITS);
  unsigned long long* thresh = (unsigned long long*)((char*)d_ws + WS_THRESH);

  k_init<<<1, 32, 0, stream>>>(counts);
  k_count<<<dim3(20, BATCH), 1024, 0, stream>>>(data, counts, bits);
  k_select<<<BATCH * 2, 256, 0, stream>>>(counts, bits, thresh, seedp);
  k_apply<<<dim3(V4_PER_ROW / 256, BATCH), 256, 0, stream>>>(
      data, (v4f*)d_out, bits, thresh, seedp);
}